// Multi_View_CL_Loss_74955769249955
// MI455X (gfx1250) — compile-verified
//
#include <hip/hip_runtime.h>

// MI455X / gfx1250 (wave32). Problem: gather + batched 128-dim f32 dots +
// log-sigmoid reduction. L2-resident tables (3 x 32MB < 192MB L2) =>
// gather-bandwidth bound; dots done in VALU (one wave per anchor, b128 row
// loads = 512B/wave, perfectly coalesced). WMMA is used where it is exactly
// correct: the final 16-partial reduction via C = A x ones (B=ones is layout
// independent; A 16x4 f32 and C 16x16 f32 layouts are documented).

typedef __attribute__((ext_vector_type(2))) float v2f;
typedef __attribute__((ext_vector_type(8))) float v8f;

#define DIMS 128
#define NSAMP 4
#define WAVES_PER_BLOCK 8
#define BLOCK_THREADS (WAVES_PER_BLOCK * 32)
#define CLIP_MIN (-70.0f)

__device__ __forceinline__ float wave_sum32(float v) {
  // wave32 butterfly reduction (warpSize == 32 on gfx1250)
  v += __shfl_xor(v, 1, 32);
  v += __shfl_xor(v, 2, 32);
  v += __shfl_xor(v, 4, 32);
  v += __shfl_xor(v, 8, 32);
  v += __shfl_xor(v, 16, 32);
  return v;
}

__device__ __forceinline__ float dot4(const float4 a, const float4 b) {
  return fmaf(a.x, b.x, fmaf(a.y, b.y, fmaf(a.z, b.z, a.w * b.w)));
}

__device__ __forceinline__ float log_sigmoid(float x) {
  // jax.nn.log_sigmoid: -softplus(-x), stable form
  return fminf(x, 0.0f) - log1pf(expf(-fabsf(x)));
}

__device__ __forceinline__ float4 ldg4(const float* __restrict__ p) {
  return *reinterpret_cast<const float4*>(p);
}

// One term: anchor (this wave's anchor row, 4 dims per lane) vs gathered rows
// of `tab`. a_close = anchor . (sum of 4 close rows); a_away_s separate.
__device__ __forceinline__ float term_loss(const float4 anch,
                                           const float* __restrict__ tab,
                                           const int4 c4, const int4 a4,
                                           const int lane) {
  const int off = lane * 4;  // 4 floats per lane, 32 lanes = 128 dims = 512B row

  // close rows: elementwise-sum first (einsum 'bd,bsd->b' sums over s too)
  const float4 c0 = ldg4(tab + (size_t)c4.x * DIMS + off);
  const float4 c1 = ldg4(tab + (size_t)c4.y * DIMS + off);
  const float4 c2 = ldg4(tab + (size_t)c4.z * DIMS + off);
  const float4 c3 = ldg4(tab + (size_t)c4.w * DIMS + off);
  float4 rc;
  rc.x = (c0.x + c1.x) + (c2.x + c3.x);
  rc.y = (c0.y + c1.y) + (c2.y + c3.y);
  rc.z = (c0.z + c1.z) + (c2.z + c3.z);
  rc.w = (c0.w + c1.w) + (c2.w + c3.w);
  const float aclose = wave_sum32(dot4(anch, rc));

  // away rows: individual dots
  const float4 w0 = ldg4(tab + (size_t)a4.x * DIMS + off);
  const float4 w1 = ldg4(tab + (size_t)a4.y * DIMS + off);
  const float4 w2 = ldg4(tab + (size_t)a4.z * DIMS + off);
  const float4 w3 = ldg4(tab + (size_t)a4.w * DIMS + off);
  const float d0 = wave_sum32(dot4(anch, w0));
  const float d1 = wave_sum32(dot4(anch, w1));
  const float d2 = wave_sum32(dot4(anch, w2));
  const float d3 = wave_sum32(dot4(anch, w3));

  float loss;
  loss  = log_sigmoid(fmaxf(aclose - (float)NSAMP * d0, CLIP_MIN));
  loss += log_sigmoid(fmaxf(aclose - (float)NSAMP * d1, CLIP_MIN));
  loss += log_sigmoid(fmaxf(aclose - (float)NSAMP * d2, CLIP_MIN));
  loss += log_sigmoid(fmaxf(aclose - (float)NSAMP * d3, CLIP_MIN));
  return loss;
}

// One wave per anchor; block = 8 waves = 8 anchors; per-block partial pair to ws.
__global__ void cl_loss_partials(const float* __restrict__ graph,
                                 const float* __restrict__ seq,
                                 const float* __restrict__ pred,
                                 const int* __restrict__ cid,
                                 const int* __restrict__ aid,
                                 float* __restrict__ partial) {
  __shared__ float l12s[WAVES_PER_BLOCK];
  __shared__ float l1s[WAVES_PER_BLOCK];

  const int lane = threadIdx.x & 31;
  const int wid  = threadIdx.x >> 5;
  const int b    = blockIdx.x * WAVES_PER_BLOCK + wid;
  const int off  = lane * 4;

  const float4 ga = ldg4(graph + (size_t)b * DIMS + off);
  const float4 sa = ldg4(seq   + (size_t)b * DIMS + off);
  const float4 pa = ldg4(pred  + (size_t)b * DIMS + off);
  const int4 c4 = *reinterpret_cast<const int4*>(cid + (size_t)b * NSAMP);
  const int4 a4 = *reinterpret_cast<const int4*>(aid + (size_t)b * NSAMP);

  // loss_1 = term(graph, seq); loss_2 = term(seq, graph); level_1 = term(pred, pred)
  const float l12 = term_loss(ga, seq, c4, a4, lane) +
                    term_loss(sa, graph, c4, a4, lane);
  const float l1  = term_loss(pa, pred, c4, a4, lane);

  if (lane == 0) { l12s[wid] = l12; l1s[wid] = l1; }
  __syncthreads();
  if (threadIdx.x == 0) {
    float s12 = 0.0f, s1 = 0.0f;
#pragma unroll
    for (int i = 0; i < WAVES_PER_BLOCK; ++i) { s12 += l12s[i]; s1 += l1s[i]; }
    partial[2 * blockIdx.x + 0] = s12;
    partial[2 * blockIdx.x + 1] = s1;
  }
}

// Single-block deterministic epilogue. Last 16 wave-partials (8x loss12 +
// 8x loss1) are reduced with V_WMMA_F32_16X16X4_F32 against an all-ones B:
// C[m,n] = sum_k A[m,k]. Rows 0-1 carry loss12, rows 2-3 carry loss1.
__global__ void cl_loss_final(const float* __restrict__ partial, int nblocks,
                              float* __restrict__ out) {
  __shared__ float s12a[WAVES_PER_BLOCK];
  __shared__ float s1a[WAVES_PER_BLOCK];

  float s12 = 0.0f, s1 = 0.0f;
  for (int i = threadIdx.x; i < nblocks; i += BLOCK_THREADS) {
    s12 += partial[2 * i + 0];
    s1  += partial[2 * i + 1];
  }
  s12 = wave_sum32(s12);
  s1  = wave_sum32(s1);

  const int lane = threadIdx.x & 31;
  const int wid  = threadIdx.x >> 5;
  if (lane == 0) { s12a[wid] = s12; s1a[wid] = s1; }
  __syncthreads();

  // Build A (16x4 f32): A[m,k] -> lane = m + 16*(k/2), vgpr = k%2 (ISA layout).
  // Row 0: s12a[0..3], row 1: s12a[4..7], row 2: s1a[0..3], row 3: s1a[4..7].
  const int r  = lane & 15;
  const int hi = lane >> 4;  // 0 => k in {0,1}; 1 => k in {2,3}
  float a0 = 0.0f, a1 = 0.0f;
  if (r == 0)      { a0 = s12a[2 * hi + 0]; a1 = s12a[2 * hi + 1]; }
  else if (r == 1) { a0 = s12a[4 + 2 * hi]; a1 = s12a[5 + 2 * hi]; }
  else if (r == 2) { a0 = s1a[2 * hi + 0];  a1 = s1a[2 * hi + 1];  }
  else if (r == 3) { a0 = s1a[4 + 2 * hi];  a1 = s1a[5 + 2 * hi];  }

  v2f A; A[0] = a0; A[1] = a1;
  v2f Bones; Bones[0] = 1.0f; Bones[1] = 1.0f;  // all-ones: layout independent
  v8f C = {};
  // All 8 waves execute with EXEC all-ones (WMMA requirement); all compute the
  // same result, only global thread 0 stores.
  C = __builtin_amdgcn_wmma_f32_16x16x4_f32(
      /*neg_a=*/false, A, /*neg_b=*/false, Bones,
      /*c_mod=*/(short)0, C, /*reuse_a=*/false, /*reuse_b=*/false);

  if (threadIdx.x == 0) {
    // Lane 0 holds column N=0: C[0..7, 0] in VGPRs 0..7.
    out[0] = C[0] + C[1];  // loss_level_2 = loss_1 + loss_2
    out[1] = C[2] + C[3];  // loss_level_1
  }
}

extern "C" void kernel_launch(void* const* d_in, const int* in_sizes, int n_in,
                              void* d_out, int out_size, void* d_ws, size_t ws_size,
                              hipStream_t stream) {
  const float* graph = (const float*)d_in[0];
  const float* seq   = (const float*)d_in[1];
  const float* pred  = (const float*)d_in[2];
  const int*   cid   = (const int*)d_in[3];
  const int*   aid   = (const int*)d_in[4];
  float* out = (float*)d_out;
  float* ws  = (float*)d_ws;

  const int B = in_sizes[3] / NSAMP;              // 65536
  const int nblocks = B / WAVES_PER_BLOCK;        // 8192 -> 64KB of ws partials

  cl_loss_partials<<<nblocks, BLOCK_THREADS, 0, stream>>>(graph, seq, pred,
                                                          cid, aid, ws);
  cl_loss_final<<<1, BLOCK_THREADS, 0, stream>>>(ws, nblocks, out);
}